// MemoryLayerWithLNResidual_29128468201512
// MI455X (gfx1250) — compile-verified
//
#include <hip/hip_runtime.h>
#include <math.h>

#define D 1024
#define SLOTS 128
#define HID 128
#define TOK 16

// LDS row pitches padded by 4 floats (16B) to avoid 16-way bank conflicts on
// A-fragment reads (row stride mod 64 banks becomes 4*r instead of 0).
#define QLDA (D + 4)        // 1028
#define FLDA (2 * D + 4)    // 2052
#define SLDA (SLOTS + 4)    // 132

typedef __attribute__((ext_vector_type(2))) float v2f;
typedef __attribute__((ext_vector_type(4))) float v4f;
typedef __attribute__((ext_vector_type(8))) float v8f;

__device__ __forceinline__ v8f wmma_f32(v2f a, v2f b, v8f c) {
  // D = A(16x4) * B(4x16) + C(16x16), fp32
  return __builtin_amdgcn_wmma_f32_16x16x4_f32(false, a, false, b, (short)0, c,
                                               false, false);
}

// ---- CDNA5 async memory->LDS copy (ASYNCcnt-tracked) --------------------
// LDS destination VGPR holds the wave-relative LDS byte address; for a
// generic pointer to shared memory that is the low 32 address bits
__device__ __forceinline__ void cp_async16(float* ldst, const float* gsrc) {
  unsigned lds_off = (unsigned)(unsigned long long)ldst;
  asm volatile("global_load_async_to_lds_b128 %0, %1, off"
               :
               : "v"(lds_off), "v"(gsrc)
               : "memory");
}
__device__ __forceinline__ void async_wait0() {
  asm volatile("s_wait_asynccnt 0x0" ::: "memory");
}

// A fragment from row-major [*, lda]: lane l -> row = row0 + (l&15),
// K pair = k0 + 2*(l>>4) (contiguous float2).
__device__ __forceinline__ v2f frag_a_rm(const float* A, int lda, int row0,
                                         int k0, int lane) {
  const float* p = A + (size_t)(row0 + (lane & 15)) * lda + (k0 + 2 * (lane >> 4));
  return v2f{p[0], p[1]};
}
// B fragment from k-major [K, ldb] (B[k][n]): lane l -> col = n0 + (l&15),
// K pair rows k0 + 2*(l>>4) (+0, +1) -> stride-ldb pair.
__device__ __forceinline__ v2f frag_b_kn(const float* B, int ldb, int k0,
                                         int n0, int lane) {
  const float* p = B + (size_t)(k0 + 2 * (lane >> 4)) * ldb + (n0 + (lane & 15));
  return v2f{p[0], p[ldb]};
}
// B fragment where B[k][n] = Bt[n*ldb + k] (transposed row-major): contiguous.
__device__ __forceinline__ v2f frag_b_nk(const float* Bt, int ldb, int k0,
                                         int n0, int lane) {
  const float* p = Bt + (size_t)(n0 + (lane & 15)) * ldb + (k0 + 2 * (lane >> 4));
  return v2f{p[0], p[1]};
}
// C/D layout: VGPR r -> M = m0 + r + 8*(l>>4), N = n0 + (l&15)
__device__ __forceinline__ void store_c(float* C, int ldc, int m0, int n0,
                                        int lane, v8f c) {
  int col = n0 + (lane & 15);
  int row = m0 + 8 * (lane >> 4);
#pragma unroll
  for (int r = 0; r < 8; ++r) C[(size_t)(row + r) * ldc + col] = c[r];
}

// -------------------------------------------------------------------------
// Kernel 1: per 16-token tile -- attention read + write-weight MLP.
//   logits = q keys^T / temp ; softmax ; mem_read = attn values
//   ww = sigmoid(gelu(q W1 + b1) W2 + b2)
// 8 waves / WG. q tile async-staged once in LDS, reused by 2 K=1024 GEMMs.
// -------------------------------------------------------------------------
extern "C" __global__ void __launch_bounds__(256)
k1_read_write(const float* __restrict__ x, const float* __restrict__ keys,
              const float* __restrict__ values, const float* __restrict__ temp,
              const float* __restrict__ W1, const float* __restrict__ b1,
              const float* __restrict__ W2, const float* __restrict__ b2,
              float* __restrict__ mem_read, float* __restrict__ ww) {
  extern __shared__ float smem[];
  float* qs    = smem;                    // 16 * QLDA
  float* attn  = qs + TOK * QLDA;         // 16 * SLDA
  float* hid   = attn + TOK * SLDA;       // 16 * SLDA
  float* red   = hid + TOK * SLDA;        // 256
  float* rstat = red + 256;               // 16

  const int tid = threadIdx.x;
  const int lane = tid & 31;
  const int wave = tid >> 5;
  const long tok0 = (long)blockIdx.x * TOK;
  const float invt = 1.0f / fmaxf(temp[0], 1e-6f);

  // async-stage q tile (16 x 1024) directly into LDS, 16B per op
  for (int i = tid; i < TOK * (D / 4); i += 256) {
    int row = i / (D / 4);
    int c4  = i % (D / 4);
    cp_async16(qs + row * QLDA + c4 * 4, x + (tok0 + row) * D + c4 * 4);
  }
  async_wait0();
  __syncthreads();

  // GEMM1: logits[16,128]; wave w owns slot tile w*16
  {
    int n0 = wave * 16;
    v8f c = {};
    for (int k0 = 0; k0 < D; k0 += 4) {
      v2f a = frag_a_rm(qs, QLDA, 0, k0, lane);
      v2f b = frag_b_nk(keys, D, k0, n0, lane);  // keys^T: contiguous in k
      c = wmma_f32(a, b, c);
    }
#pragma unroll
    for (int r = 0; r < 8; ++r) c[r] *= invt;
    store_c(attn, SLDA, 0, n0, lane, c);
  }
  __syncthreads();

  // softmax over SLOTS per token row: 16 threads per row, 8 cols per thread
  {
    int row = tid >> 4, sub = tid & 15;
    float m = -3.4e38f;
    for (int c = 0; c < 8; ++c) m = fmaxf(m, attn[row * SLDA + sub * 8 + c]);
    red[row * 16 + sub] = m;
    __syncthreads();
    if (sub == 0) {
      float mm = red[row * 16];
      for (int s = 1; s < 16; ++s) mm = fmaxf(mm, red[row * 16 + s]);
      rstat[row] = mm;
    }
    __syncthreads();
    float mm = rstat[row];
    float ps = 0.f;
    for (int c = 0; c < 8; ++c) {
      float e = __expf(attn[row * SLDA + sub * 8 + c] - mm);
      attn[row * SLDA + sub * 8 + c] = e;
      ps += e;
    }
    __syncthreads();
    red[row * 16 + sub] = ps;
    __syncthreads();
    if (sub == 0) {
      float s = 0.f;
      for (int ss = 0; ss < 16; ++ss) s += red[row * 16 + ss];
      rstat[row] = 1.0f / s;
    }
    __syncthreads();
    float inv = rstat[row];
    for (int c = 0; c < 8; ++c) attn[row * SLDA + sub * 8 + c] *= inv;
  }
  __syncthreads();

  // GEMM2: mem_read[16,1024] = attn @ values; wave w covers cols [w*128,+128)
  // two N-tiles per K-loop -> one shared A fragment per step
  for (int j = 0; j < 4; ++j) {
    int n0 = wave * 128 + j * 32;
    v8f c0 = {}, c1 = {};
    for (int k0 = 0; k0 < SLOTS; k0 += 4) {
      v2f a  = frag_a_rm(attn, SLDA, 0, k0, lane);
      v2f b0 = frag_b_kn(values, D, k0, n0, lane);
      v2f b1 = frag_b_kn(values, D, k0, n0 + 16, lane);
      c0 = wmma_f32(a, b0, c0);
      c1 = wmma_f32(a, b1, c1);
    }
    int col = n0 + (lane & 15);
    int rb = 8 * (lane >> 4);
#pragma unroll
    for (int r = 0; r < 8; ++r) {
      mem_read[(tok0 + rb + r) * D + col] = c0[r];
      mem_read[(tok0 + rb + r) * D + col + 16] = c1[r];
    }
  }

  // GEMM3: hid[16,128] = gelu(q W1 + b1); wave w -> n0 = w*16
  {
    int n0 = wave * 16;
    v8f c = {};
    for (int k0 = 0; k0 < D; k0 += 4) {
      v2f a = frag_a_rm(qs, QLDA, 0, k0, lane);
      v2f b = frag_b_kn(W1, HID, k0, n0, lane);
      c = wmma_f32(a, b, c);
    }
    int col = n0 + (lane & 15);
    int rb = 8 * (lane >> 4);
#pragma unroll
    for (int r = 0; r < 8; ++r) {
      float v = c[r] + b1[col];
      v = 0.5f * v * (1.0f + erff(v * 0.70710678118654752f));  // exact gelu
      hid[(rb + r) * SLDA + col] = v;
    }
  }
  __syncthreads();

  // GEMM4: ww[16,128] = sigmoid(hid W2 + b2); wave w -> n0 = w*16
  {
    int n0 = wave * 16;
    v8f c = {};
    for (int k0 = 0; k0 < HID; k0 += 4) {
      v2f a = frag_a_rm(hid, SLDA, 0, k0, lane);
      v2f b = frag_b_kn(W2, SLOTS, k0, n0, lane);
      c = wmma_f32(a, b, c);
    }
    int col = n0 + (lane & 15);
    int rb = 8 * (lane >> 4);
#pragma unroll
    for (int r = 0; r < 8; ++r) {
      float v = c[r] + b2[col];
      ww[(tok0 + rb + r) * SLOTS + col] = 1.0f / (1.0f + __expf(-v));
    }
  }
}

// -------------------------------------------------------------------------
// Kernel: new_values[128,1024] = values + 0.1 * ww^T @ q
// Workgroup split-K: 8 waves each cover Ntok/8 tokens for one 16x16 tile,
// partial C tiles reduced through LDS in fixed wave order (deterministic).
// -------------------------------------------------------------------------
extern "C" __global__ void __launch_bounds__(256)
k_write(const float* __restrict__ ww, const float* __restrict__ x,
        const float* __restrict__ values, float* __restrict__ newv, int Ntok) {
  __shared__ float acc[8 * 256];
  const int tid = threadIdx.x, lane = tid & 31, wave = tid >> 5;
  const int m0 = blockIdx.x * 16;  // slot rows
  const int n0 = blockIdx.y * 16;  // d cols
  const int kchunk = Ntok >> 3;    // tokens per wave
  const int kbeg = wave * kchunk;
  v8f c = {};
  for (int k0 = kbeg; k0 < kbeg + kchunk; k0 += 4) {
    v2f a = frag_b_kn(ww, SLOTS, k0, m0, lane);  // A fragment (ww is k-major)
    v2f b = frag_b_kn(x, D, k0, n0, lane);
    c = wmma_f32(a, b, c);
  }
  store_c(acc + wave * 256, 16, 0, 0, lane, c);
  __syncthreads();
  int r = tid >> 4, cc = tid & 15;
  float s = 0.f;
#pragma unroll
  for (int w = 0; w < 8; ++w) s += acc[w * 256 + r * 16 + cc];
  size_t idx = (size_t)(m0 + r) * D + (n0 + cc);
  newv[idx] = values[idx] + 0.1f * s;
}

// -------------------------------------------------------------------------
// Kernel: fused[N,1024] = concat(x, mem_read) @ Wf + bf  (K = 2048)
// 16 tokens/WG; concat tile async-staged in LDS, 8 waves x 4x2 n-tiles.
// -------------------------------------------------------------------------
extern "C" __global__ void __launch_bounds__(256)
k_fuse(const float* __restrict__ x, const float* __restrict__ mem_read,
       const float* __restrict__ Wf, const float* __restrict__ bf,
       float* __restrict__ fused) {
  extern __shared__ float smem[];  // 16 * FLDA
  const int tid = threadIdx.x, lane = tid & 31, wave = tid >> 5;
  const long tok0 = (long)blockIdx.x * TOK;
  const int K2 = 2 * D;
  for (int i = tid; i < TOK * (K2 / 4); i += 256) {
    int row = i / (K2 / 4);
    int c4  = i % (K2 / 4);
    const float* src = (c4 < D / 4)
                           ? (x + (tok0 + row) * D + c4 * 4)
                           : (mem_read + (tok0 + row) * D + (c4 - D / 4) * 4);
    cp_async16(smem + row * FLDA + c4 * 4, src);
  }
  async_wait0();
  __syncthreads();
  for (int j = 0; j < 4; ++j) {
    int n0 = wave * 128 + j * 32;
    v8f c0 = {}, c1 = {};
    for (int k0 = 0; k0 < K2; k0 += 4) {
      v2f a  = frag_a_rm(smem, FLDA, 0, k0, lane);
      v2f b0 = frag_b_kn(Wf, D, k0, n0, lane);
      v2f b1 = frag_b_kn(Wf, D, k0, n0 + 16, lane);
      c0 = wmma_f32(a, b0, c0);
      c1 = wmma_f32(a, b1, c1);
    }
    int col = n0 + (lane & 15);
    int rb = 8 * (lane >> 4);
#pragma unroll
    for (int r = 0; r < 8; ++r) {
      fused[(tok0 + rb + r) * D + col] = c0[r] + bf[col];
      fused[(tok0 + rb + r) * D + col + 16] = c1[r] + bf[col + 16];
    }
  }
}

// -------------------------------------------------------------------------
// Kernel: out_pre[N,1024] = ln1(fused) @ Wo + bo + x   (residual pre-LN2)
// -------------------------------------------------------------------------
extern "C" __global__ void __launch_bounds__(256)
k_out(const float* __restrict__ fused_ln, const float* __restrict__ Wo,
      const float* __restrict__ bo, const float* __restrict__ x,
      float* __restrict__ out_pre) {
  extern __shared__ float smem[];  // 16 * QLDA
  const int tid = threadIdx.x, lane = tid & 31, wave = tid >> 5;
  const long tok0 = (long)blockIdx.x * TOK;
  for (int i = tid; i < TOK * (D / 4); i += 256) {
    int row = i / (D / 4);
    int c4  = i % (D / 4);
    cp_async16(smem + row * QLDA + c4 * 4, fused_ln + (tok0 + row) * D + c4 * 4);
  }
  async_wait0();
  __syncthreads();
  for (int j = 0; j < 4; ++j) {
    int n0 = wave * 128 + j * 32;
    v8f c0 = {}, c1 = {};
    for (int k0 = 0; k0 < D; k0 += 4) {
      v2f a  = frag_a_rm(smem, QLDA, 0, k0, lane);
      v2f b0 = frag_b_kn(Wo, D, k0, n0, lane);
      v2f b1 = frag_b_kn(Wo, D, k0, n0 + 16, lane);
      c0 = wmma_f32(a, b0, c0);
      c1 = wmma_f32(a, b1, c1);
    }
    int col = n0 + (lane & 15);
    int rb = 8 * (lane >> 4);
#pragma unroll
    for (int r = 0; r < 8; ++r) {
      size_t i0 = (tok0 + rb + r) * D + col;
      out_pre[i0] = c0[r] + bo[col] + x[i0];
      out_pre[i0 + 16] = c1[r] + bo[col + 16] + x[i0 + 16];
    }
  }
}

// -------------------------------------------------------------------------
// Row LayerNorm (eps = 1e-5, biased variance). One 256-thread WG per row.
// -------------------------------------------------------------------------
extern "C" __global__ void __launch_bounds__(256)
k_ln(const float* __restrict__ in, const float* __restrict__ g,
     const float* __restrict__ b, float* __restrict__ out) {
  __shared__ float red[256];
  const long row = blockIdx.x;
  const float* p = in + row * D;
  float s = 0.f, s2 = 0.f;
  for (int i = threadIdx.x; i < D; i += 256) {
    float v = p[i];
    s += v;
    s2 += v * v;
  }
  red[threadIdx.x] = s;
  __syncthreads();
  for (int off = 128; off > 0; off >>= 1) {
    if (threadIdx.x < off) red[threadIdx.x] += red[threadIdx.x + off];
    __syncthreads();
  }
  float mean = red[0] * (1.0f / D);
  __syncthreads();
  red[threadIdx.x] = s2;
  __syncthreads();
  for (int off = 128; off > 0; off >>= 1) {
    if (threadIdx.x < off) red[threadIdx.x] += red[threadIdx.x + off];
    __syncthreads();
  }
  float var = red[0] * (1.0f / D) - mean * mean;
  float rstd = rsqrtf(var + 1e-5f);
  for (int i = threadIdx.x; i < D; i += 256)
    out[row * D + i] = (p[i] - mean) * rstd * g[i] + b[i];
}

// -------------------------------------------------------------------------
extern "C" void kernel_launch(void* const* d_in, const int* in_sizes, int n_in,
                              void* d_out, int out_size, void* d_ws,
                              size_t ws_size, hipStream_t stream) {
  const float* x           = (const float*)d_in[0];
  const float* keys        = (const float*)d_in[1];
  const float* values      = (const float*)d_in[2];
  const float* temperature = (const float*)d_in[3];
  const float* W1          = (const float*)d_in[4];
  const float* b1          = (const float*)d_in[5];
  const float* W2          = (const float*)d_in[6];
  const float* b2          = (const float*)d_in[7];
  const float* Wf          = (const float*)d_in[8];
  const float* bf          = (const float*)d_in[9];
  const float* ln1_g       = (const float*)d_in[10];
  const float* ln1_b       = (const float*)d_in[11];
  const float* Wo          = (const float*)d_in[12];
  const float* bo          = (const float*)d_in[13];
  const float* ln2_g       = (const float*)d_in[14];
  const float* ln2_b       = (const float*)d_in[15];

  const int N = in_sizes[0] / D;  // 32768 tokens
  float* out  = (float*)d_out;
  float* newv = out + (size_t)N * D;  // new_values output region

  // Workspace: [mem_read | out_pre: N*D] [ww (then fused overlays): N*D]
  float* ws       = (float*)d_ws;
  float* mem_read = ws;                  // N*D, reused as out_pre later
  float* ww       = ws + (size_t)N * D;  // N*SLOTS (consumed by k_write)
  float* fused    = ws + (size_t)N * D;  // aliases ww after k_write done

  const size_t lds1 = (size_t)(TOK * QLDA + 2 * TOK * SLDA + 256 + 16) *
                      sizeof(float);
  k1_read_write<<<N / TOK, 256, lds1, stream>>>(x, keys, values, temperature,
                                                W1, b1, W2, b2, mem_read, ww);
  k_write<<<dim3(SLOTS / 16, D / 16), 256, 0, stream>>>(ww, x, values, newv, N);
  k_fuse<<<N / TOK, 256, (size_t)TOK * FLDA * sizeof(float), stream>>>(
      x, mem_read, Wf, bf, fused);
  k_ln<<<N, 256, 0, stream>>>(fused, ln1_g, ln1_b, fused);
  k_out<<<N / TOK, 256, (size_t)TOK * QLDA * sizeof(float), stream>>>(
      fused, Wo, bo, x, mem_read /* out_pre */);
  k_ln<<<N, 256, 0, stream>>>(mem_read, ln2_g, ln2_b, out);
}